// MultiScaleRoIAlign_66228395704544
// MI455X (gfx1250) — compile-verified
//
#include <hip/hip_runtime.h>
#include <hip/hip_bf16.h>

// MultiScaleRoIAlign for MI455X (gfx1250, wave32).
//
// Per (roi, output-row iy) workgroup:
//   1) gather the 14 x-tap columns into LDS. Preferred path: async VMEM->LDS
//      (GLOBAL_LOAD_ASYNC_TO_LDS_B128, ASYNCcnt) storing both raw y-rows;
//      fallback path: regular loads + y-blend + ds_store.
//   2) out[ix, c] = sum_s Wx[ix,s] * r[s,c]  via V_WMMA_F32_16X16X4_F32
//      (M=16 pad of 7 ix rows, K=16 pad of 14 taps in 4 chunks, N=256 in 16 chunks)
// Memory traffic identical to a direct 4-corner gather (~0.5 GB total,
// ~21us at 23.3 TB/s; pyramid fits in 192MB L2), WMMA compute is negligible.

typedef __attribute__((ext_vector_type(2))) float v2f;
typedef __attribute__((ext_vector_type(8))) float v8f;
typedef __attribute__((ext_vector_type(4))) int   v4i;

#define CROP    7
#define CDIM    256
#define SPITCH  272   // LDS row pitch in floats; 272 % 64 == 16 -> conflict-free B reads
#define NPERB   1000

#if defined(__gfx1250__) && __has_builtin(__builtin_amdgcn_global_load_async_to_lds_b128)
#define USE_ASYNC_LDS 1
#define RROWS 32      // rows (s,j): row = 2*s + j, s=0..15 (s>=14 zero), j = y-tap
#else
#define USE_ASYNC_LDS 0
#define RROWS 16      // rows s = 0..15 of pre-blended data (s>=14 zero)
#endif

__global__ __launch_bounds__(128) void roialign_wmma_kernel(
    const float* __restrict__ bboxes,
    const float* __restrict__ f0, const float* __restrict__ f1,
    const float* __restrict__ f2, const float* __restrict__ f3,
    const float* __restrict__ f4,
    float* __restrict__ out)
{
    __shared__ float rbuf[RROWS * SPITCH];

    const int roi = blockIdx.x;   // 0..1999
    const int iy  = blockIdx.y;   // 0..6
    const int tid = threadIdx.x;  // 0..127
    const int b   = roi / NPERB;

    // ---- box -> normalized clipped coords (bbox layout: x1,y1,x2,y2) ----
    const float bx1 = bboxes[roi * 4 + 0];
    const float by1 = bboxes[roi * 4 + 1];
    const float bx2 = bboxes[roi * 4 + 2];
    const float by2 = bboxes[roi * 4 + 3];
    const float inv = 1.0f / 1024.0f;
    const float x1n = fminf(fmaxf(bx1, 0.0f), 1024.0f) * inv;
    const float y1n = fminf(fmaxf(by1, 0.0f), 1024.0f) * inv;
    const float x2n = fminf(fmaxf(bx2, 0.0f), 1024.0f) * inv;
    const float y2n = fminf(fmaxf(by2, 0.0f), 1024.0f) * inv;

    // ---- FPN level: round-half-even(log2(sqrt(h*w) * 1024/56)), clip [0,4] ----
    const float h  = y2n - y1n;
    const float w  = x2n - x1n;
    const float lv = log2f(sqrtf(h * w) * (1024.0f / 56.0f));
    const int level = (int)fminf(fmaxf(rintf(lv), 0.0f), 4.0f);

    int S; const float* F;
    switch (level) {
        case 0:  S = 256; F = f0; break;
        case 1:  S = 128; F = f1; break;
        case 2:  S = 64;  F = f2; break;
        case 3:  S = 32;  F = f3; break;
        default: S = 16;  F = f4; break;
    }
    F += (size_t)b * (size_t)S * (size_t)S * CDIM;

    const float Sm1    = (float)(S - 1);
    const float yscale = (y2n - y1n) * Sm1 * (1.0f / 6.0f);
    const float xscale = (x2n - x1n) * Sm1 * (1.0f / 6.0f);

    // ---- y sample for this output row ----
    const float in_y = y1n * Sm1 + (float)iy * yscale;
    const float fy   = floorf(in_y);
    const float wy   = in_y - fy;
    int y0 = (int)fy;            y0  = min(max(y0, 0), S - 1);
    int y1i = min(max(y0 + 1, 0), S - 1);
    const bool valid_y = (in_y >= 0.0f) && (in_y <= Sm1);
    const float omwy = 1.0f - wy;

#if USE_ASYNC_LDS
    // ---- async-stage raw rows: rbuf[2*s + j][c] = F[yj, x_s, c] ----
    for (int t = tid; t < 28 * (CDIM / 4); t += 128) {     // 14 taps x 2 y-rows
        const int row = t >> 6;          // 0..27
        const int c4  = (t & 63) << 2;
        const int s   = row >> 1;
        const int j   = row & 1;
        const int ixs = s >> 1;
        const int tap = s & 1;
        const float in_x = x1n * Sm1 + (float)ixs * xscale;
        int x = (int)floorf(in_x) + tap;
        x = min(max(x, 0), S - 1);
        const int yy = j ? y1i : y0;
        const float* src = &F[((size_t)yy * S + x) * CDIM + c4];
        __builtin_amdgcn_global_load_async_to_lds_b128(
            (__attribute__((address_space(1))) v4i*)(src),
            (__attribute__((address_space(3))) v4i*)(&rbuf[row * SPITCH + c4]),
            0, 0);
    }
    // zero the K-padding rows (s = 14,15 -> rows 28..31)
    for (int t = tid; t < 4 * (CDIM / 4); t += 128) {
        const int row = 28 + (t >> 6);
        const int c4  = (t & 63) << 2;
        float4 z; z.x = 0.f; z.y = 0.f; z.z = 0.f; z.w = 0.f;
        *(float4*)&rbuf[row * SPITCH + c4] = z;
    }
#if __has_builtin(__builtin_amdgcn_s_wait_asynccnt)
    __builtin_amdgcn_s_wait_asynccnt(0);
#else
    asm volatile("s_wait_asynccnt 0x0" ::: "memory");
#endif
#else
    // ---- fallback: stage r[s][c] = (1-wy)*F[y0,xs,c] + wy*F[y1i,xs,c] ----
    for (int t = tid; t < 16 * (CDIM / 4); t += 128) {
        const int s  = t >> 6;          // 0..15
        const int c4 = (t & 63) << 2;
        float4 r;
        if (s < 14) {
            const int ixs = s >> 1;
            const int tap = s & 1;
            const float in_x = x1n * Sm1 + (float)ixs * xscale;
            int x = (int)floorf(in_x) + tap;
            x = min(max(x, 0), S - 1);
            const float4 v0 = *(const float4*)&F[((size_t)y0  * S + x) * CDIM + c4];
            const float4 v1 = *(const float4*)&F[((size_t)y1i * S + x) * CDIM + c4];
            r.x = omwy * v0.x + wy * v1.x;
            r.y = omwy * v0.y + wy * v1.y;
            r.z = omwy * v0.z + wy * v1.z;
            r.w = omwy * v0.w + wy * v1.w;
        } else {
            r.x = 0.0f; r.y = 0.0f; r.z = 0.0f; r.w = 0.0f;
        }
        *(float4*)&rbuf[s * SPITCH + c4] = r;
    }
#endif
    __syncthreads();

    // ---- WMMA phase: out = Wx(7x14, padded 16x16) x r(16x256) ----
    const int wave = tid >> 5;
    const int lane = tid & 31;
    const int half = lane >> 4;   // 0: lanes 0-15, 1: lanes 16-31
    const int m    = lane & 15;   // A-row (ix) / D-column (channel in chunk)

    // per-lane x weights for A row m (only s==2m, 2m+1 are non-zero)
    float wxm = 0.0f;
    bool  vm  = false;
    {
        const float in_x = x1n * Sm1 + (float)m * xscale;
        wxm = in_x - floorf(in_x);
        vm  = (m < CROP) && valid_y && (in_x >= 0.0f) && (in_x <= Sm1);
    }

    const size_t outbase = ((size_t)roi * 49 + (size_t)iy * CROP) * CDIM;

    for (int nc = wave * 4; nc < wave * 4 + 4; ++nc) {
        const int nb = nc * 16;
        v8f acc = {0.f, 0.f, 0.f, 0.f, 0.f, 0.f, 0.f, 0.f};
        #pragma unroll
        for (int kc = 0; kc < 4; ++kc) {
            // A (16x4 f32): VGPR0 <-> K = kc*4 + 2*half, VGPR1 <-> K+1
            const int s0 = kc * 4 + 2 * half;
            float a0 = 0.0f, a1 = 0.0f;
            if (vm) {
                const int d0 = s0 - 2 * m;
                a0 = (d0 == 0) ? (1.0f - wxm) : ((d0 == 1) ? wxm : 0.0f);
                const int d1 = d0 + 1;
                a1 = (d1 == 0) ? (1.0f - wxm) : ((d1 == 1) ? wxm : 0.0f);
            }
            // B (4x16 f32): VGPR0 <-> K = kc*4 + half, VGPR1 <-> K+2; N = m across lanes
            const int sb0 = kc * 4 + half;
            const int sb1 = kc * 4 + 2 + half;
#if USE_ASYNC_LDS
            const float b0 = omwy * rbuf[(2 * sb0)     * SPITCH + nb + m]
                           + wy   * rbuf[(2 * sb0 + 1) * SPITCH + nb + m];
            const float b1 = omwy * rbuf[(2 * sb1)     * SPITCH + nb + m]
                           + wy   * rbuf[(2 * sb1 + 1) * SPITCH + nb + m];
#else
            const float b0 = rbuf[sb0 * SPITCH + nb + m];
            const float b1 = rbuf[sb1 * SPITCH + nb + m];
#endif
            v2f A  = {a0, a1};
            v2f Bv = {b0, b1};
            acc = __builtin_amdgcn_wmma_f32_16x16x4_f32(
                false, A, false, Bv, (short)0, acc, false, false);
        }
        // D (16x16 f32): lanes 0-15 hold M = 0..7 in acc[0..7]; store rows 0..6
        if (half == 0) {
            #pragma unroll
            for (int v = 0; v < CROP; ++v) {
                out[outbase + (size_t)v * CDIM + nb + m] = acc[v];
            }
        }
    }
}

extern "C" void kernel_launch(void* const* d_in, const int* in_sizes, int n_in,
                              void* d_out, int out_size, void* d_ws, size_t ws_size,
                              hipStream_t stream) {
    const float* bboxes = (const float*)d_in[0];
    const float* f0 = (const float*)d_in[1];
    const float* f1 = (const float*)d_in[2];
    const float* f2 = (const float*)d_in[3];
    const float* f3 = (const float*)d_in[4];
    const float* f4 = (const float*)d_in[5];
    float* out = (float*)d_out;

    dim3 grid(2000, 7, 1);   // (roi, output row)
    roialign_wmma_kernel<<<grid, 128, 0, stream>>>(bboxes, f0, f1, f2, f3, f4, out);
}